// WignerEnhancedAttention_4750233829524
// MI455X (gfx1250) — compile-verified
//
#include <hip/hip_runtime.h>
#include <math.h>

typedef float v2f __attribute__((ext_vector_type(2)));
typedef float v8f __attribute__((ext_vector_type(8)));

#define B_   2
#define N_   1024
#define HG_  16
#define D_   64
#define DE_  128
#define NT_  (B_ * N_)          // 2048 (b,n) tokens
#define ROWS_ (B_ * N_ * HG_)   // 32768 projection rows

// f32 WMMA: D = A(16x4) * B(4x16) + C(16x16), all fp32 (exact vs reference)
static __device__ __forceinline__ v8f wmma_f32(v2f a, v2f b, v8f c) {
  return __builtin_amdgcn_wmma_f32_16x16x4_f32(false, a, false, b, (short)0, c, false, false);
}

// ---------------------------------------------------------------------------
// Kernel 1: wigner row-vector w[b,n,0:64] (closed form of R_vec @ T.T row d=0)
// ---------------------------------------------------------------------------
__global__ void wvec_kernel(const float* __restrict__ rot, float* __restrict__ wv) {
  int tok = blockIdx.x * blockDim.x + threadIdx.x;
  if (tok >= NT_) return;
  const float* R = rot + (size_t)tok * 9;
  float r[9];
#pragma unroll
  for (int i = 0; i < 9; ++i) r[i] = R[i];
  float w[64];
#pragma unroll
  for (int i = 0; i < 64; ++i) w[i] = 0.f;
  // l=0
  w[0] = (r[0] + r[4] + r[8]) * (1.f / 3.f);
  // l=1: identity copy of R
#pragma unroll
  for (int i = 0; i < 9; ++i) w[1 + i] = r[i];
  // l=2 diagonal
  w[10] = r[0]; w[16] = r[4]; w[22] = r[8];
  // l=2 symmetric rules
  float s01 = 0.5f * r[1] + 0.5f * r[3];                             // (0,1)
  w[11] = s01; w[15] = s01;
  float s02 = 0.7f * r[2] + 0.3f * r[6];                             // (0,2)
  w[12] = s02; w[20] = s02;
  float s12 = 0.6f * r[5] + 0.4f * r[7];                             // (1,2)
  w[17] = s12; w[21] = s12;
  float s04 = 0.3f * r[1] + 0.3f * r[3] + 0.2f * r[2] + 0.2f * r[6]; // (0,4)
  w[14] = s04; w[30] = s04;
  float* o = wv + (size_t)tok * 64;
#pragma unroll
  for (int i = 0; i < 64; ++i) o[i] = w[i];
}

// ---------------------------------------------------------------------------
// Kernel 2: fused QKV projection. One wave per token (4 tokens / block): the
// 16 rows of a tile are the 16 hg slots of one (b,n) token (contiguous 16x64
// block of x). q,k -> [b,hg,n,0:64] of the 128-wide extended buffers;
// v -> transposed vT[b,hg,dim,key].
// ---------------------------------------------------------------------------
__global__ __launch_bounds__(128) void qkv_kernel(
    const float* __restrict__ x,
    const float* __restrict__ Wq, const float* __restrict__ bq,
    const float* __restrict__ Wk, const float* __restrict__ bk,
    const float* __restrict__ Wv, const float* __restrict__ bv,
    float* __restrict__ qe, float* __restrict__ ke, float* __restrict__ vT) {
  int wid = threadIdx.x >> 5;
  int tok = blockIdx.x * 4 + wid;       // b*N + n
  int b = tok >> 10, n = tok & (N_ - 1);
  int lane = threadIdx.x & 31;
  int lm = lane & 15, lg = lane >> 4;

  const float* xb = x + (size_t)tok * HG_ * D_;   // 16 rows x 64
  v2f A[16];                                       // A-layout, K=64
#pragma unroll
  for (int kk = 0; kk < 16; ++kk) {
    int k0 = 4 * kk + 2 * lg;
    A[kk] = *(const v2f*)(xb + lm * D_ + k0);
  }

  const float* Ws[3] = {Wq, Wk, Wv};
  const float* Bs[3] = {bq, bk, bv};
#pragma unroll
  for (int s = 0; s < 3; ++s) {
    const float* W = Ws[s];
    const float* bias = Bs[s];
#pragma unroll
    for (int jt = 0; jt < 4; ++jt) {
      int j = jt * 16 + lm;
      float bv0 = bias[j];
      v8f acc = {};
#pragma unroll
      for (int r = 0; r < 8; ++r) acc[r] = bv0;   // bias broadcast (per column)
#pragma unroll
      for (int kk = 0; kk < 16; ++kk) {
        int k0 = 4 * kk + 2 * lg;
        v2f Bb = *(const v2f*)(W + (size_t)j * D_ + k0);  // out = x @ W^T
        acc = wmma_f32(A[kk], Bb, acc);
      }
#pragma unroll
      for (int r = 0; r < 8; ++r) {
        int hg = r + 8 * lg;                      // tile row == hg slot
        if (s == 0)
          qe[(((size_t)(b * HG_ + hg)) * N_ + n) * DE_ + j] = acc[r];
        else if (s == 1)
          ke[(((size_t)(b * HG_ + hg)) * N_ + n) * DE_ + j] = acc[r];
        else
          vT[(((size_t)(b * HG_ + hg)) * D_ + j) * N_ + n] = acc[r];
      }
    }
  }
}

// ---------------------------------------------------------------------------
// Kernel 3: rank-1 Wigner extension columns 64..127 of qe/ke.
//   qe_ext = fw * q0 * w ;  ke_ext = k0 * w   (so qe.ke == qk + fw*wigner)
// ---------------------------------------------------------------------------
__global__ void extend_kernel(float* __restrict__ qe, float* __restrict__ ke,
                              const float* __restrict__ wv,
                              const float* __restrict__ fw) {
  int row = blockIdx.x;                 // (b*HG+hg)*N + n
  int e = threadIdx.x;                  // 0..63
  int n = row & (N_ - 1);
  int b = row >> 14;                    // / (HG_*N_)
  float* qrow = qe + (size_t)row * DE_;
  float* krow = ke + (size_t)row * DE_;
  float q0 = qrow[0];
  float k0 = krow[0];
  float w = wv[((size_t)(b * N_ + n)) * 64 + e];
  float f = fw[0];
  qrow[64 + e] = f * q0 * w;
  krow[64 + e] = k0 * w;
}

// ---------------------------------------------------------------------------
// Kernel 4: flash attention. 4 waves per workgroup, each wave owns one
// 16-query tile; all 4 waves stream the SAME key/value tiles in lock-step so
// K/V reads hit the WGP$/L2 (4x traffic reduction vs 1 wave/WG).
// S = qe * ke^T (K=128) accumulated in TWO independent WMMA chains (even/odd
// K-steps) so the scheduler can interleave them; online softmax; O += P*V
// (P via private LDS slice, C->A layout conversion, same-wave DS in-order).
// ---------------------------------------------------------------------------
__global__ __launch_bounds__(128) void attn_kernel(
    const float* __restrict__ qe, const float* __restrict__ ke,
    const float* __restrict__ vT, const unsigned char* __restrict__ mask,
    float* __restrict__ ao) {
  __shared__ float Plds[4 * 16 * 18];   // per-wave slice; stride 18 = no bank conflicts
  int bh = blockIdx.x;                  // b*HG + hg  (0..31)
  int wid = threadIdx.x >> 5;
  int qt = blockIdx.y * 4 + wid;        // query tile (0..63)
  int b = bh >> 4;
  int lane = threadIdx.x & 31;
  int lm = lane & 15, lg = lane >> 4;
  float* Pl = &Plds[wid * 16 * 18];

  // A = 16x128 query tile (A-layout), fold 1/sqrt(D)=1/8 once here
  const float* qbase = qe + (((size_t)bh * N_) + (size_t)qt * 16) * DE_;
  v2f A[32];
#pragma unroll
  for (int kk = 0; kk < 32; ++kk) {
    int k0 = 4 * kk + 2 * lg;
    v2f a = *(const v2f*)(qbase + lm * DE_ + k0);
    a.x *= 0.125f; a.y *= 0.125f;
    A[kk] = a;
  }

  float mrow[8], lrow[8];
  v8f O[4];
#pragma unroll
  for (int r = 0; r < 8; ++r) { mrow[r] = -1e30f; lrow[r] = 0.f; }
#pragma unroll
  for (int t = 0; t < 4; ++t) O[t] = (v8f){};

  const float* kbase0 = ke + (size_t)bh * N_ * DE_;
  const float* vbase0 = vT + (size_t)bh * D_ * N_;
  const unsigned char* mb = mask + (size_t)b * N_;

  for (int kt = 0; kt < 64; ++kt) {
    const float* kbase = kbase0 + (size_t)kt * 16 * DE_;
    // prefetch next K tile and this V tile (global_prefetch_b8)
    if (kt + 1 < 64) __builtin_prefetch(kbase + 16 * DE_ + lane * 64, 0, 0);
    __builtin_prefetch(vbase0 + (size_t)(lane * 2) * N_ + kt * 16, 0, 0);

    // ---- S = (qe/8) * ke^T : 32 wmma in two independent chains ----
    v8f S0 = (v8f){};
    v8f S1 = (v8f){};
#pragma unroll
    for (int kk = 0; kk < 16; ++kk) {
      int ka = 8 * kk + 2 * lg;         // even step
      int kb = 8 * kk + 4 + 2 * lg;     // odd step
      v2f Ba = *(const v2f*)(kbase + lm * DE_ + ka);
      v2f Bb = *(const v2f*)(kbase + lm * DE_ + kb);
      S0 = wmma_f32(A[2 * kk], Ba, S0);
      S1 = wmma_f32(A[2 * kk + 1], Bb, S1);
    }
    v8f S = S0 + S1;
    // key mask (column n = lane%16)
    if (!mb[kt * 16 + lm]) {
#pragma unroll
      for (int r = 0; r < 8; ++r) S[r] = -INFINITY;
    }
    // ---- online softmax ----
    float alpha[8];
#pragma unroll
    for (int r = 0; r < 8; ++r) {
      float v = S[r];                   // row m=r+8*lg lives on 16 lanes
      v = fmaxf(v, __shfl_xor(v, 1, 32));
      v = fmaxf(v, __shfl_xor(v, 2, 32));
      v = fmaxf(v, __shfl_xor(v, 4, 32));
      v = fmaxf(v, __shfl_xor(v, 8, 32));
      float mn = fmaxf(mrow[r], v);
      alpha[r] = __expf(mrow[r] - mn);
      mrow[r] = mn;
    }
#pragma unroll
    for (int r = 0; r < 8; ++r) {
      float p = __expf(S[r] - mrow[r]);
      S[r] = p;
      float ssum = p;
      ssum += __shfl_xor(ssum, 1, 32);
      ssum += __shfl_xor(ssum, 2, 32);
      ssum += __shfl_xor(ssum, 4, 32);
      ssum += __shfl_xor(ssum, 8, 32);
      lrow[r] = lrow[r] * alpha[r] + ssum;
    }
    // P: C-layout -> private LDS slice, row-major [m][k]
    // (same-wave LDS ops are in-order; no barrier needed, slices are private)
#pragma unroll
    for (int r = 0; r < 8; ++r) Pl[(r + 8 * lg) * 18 + lm] = S[r];
    // rescale running output, then O += P*V (4 independent out-tile chains)
#pragma unroll
    for (int t = 0; t < 4; ++t)
#pragma unroll
      for (int r = 0; r < 8; ++r) O[t][r] *= alpha[r];
    const float* vb = vbase0 + kt * 16;
#pragma unroll
    for (int kk = 0; kk < 4; ++kk) {
      int k0 = 4 * kk + 2 * lg;
      v2f Pa = *(const v2f*)&Pl[lm * 18 + k0];    // A-operand from LDS
#pragma unroll
      for (int t = 0; t < 4; ++t) {
        v2f Bb = *(const v2f*)(vb + (size_t)(t * 16 + lm) * N_ + k0);
        O[t] = wmma_f32(Pa, Bb, O[t]);
      }
    }
  }

  // normalize and store attention output ao[b,hg,q,dim]
#pragma unroll
  for (int r = 0; r < 8; ++r) {
    float inv = 1.0f / lrow[r];
    int m = r + 8 * lg;
#pragma unroll
    for (int t = 0; t < 4; ++t) {
      ao[(((size_t)bh * N_) + qt * 16 + m) * D_ + t * 16 + lm] = O[t][r] * inv;
    }
  }
}

// ---------------------------------------------------------------------------
// Kernel 5: output projection out[b,n,hg,:] = ao[b,hg,n,:] @ Wo^T + bo
// (4 tokens per block, one wave each)
// ---------------------------------------------------------------------------
__global__ __launch_bounds__(128) void oproj_kernel(
    const float* __restrict__ ao, const float* __restrict__ Wo,
    const float* __restrict__ bo, float* __restrict__ out) {
  int wid = threadIdx.x >> 5;
  int tok = blockIdx.x * 4 + wid;       // b*N + n
  int b = tok >> 10, n = tok & (N_ - 1);
  int lane = threadIdx.x & 31;
  int lm = lane & 15, lg = lane >> 4;

  v2f A[16];                            // rows m==hg, gathered across heads
#pragma unroll
  for (int kk = 0; kk < 16; ++kk) {
    int k0 = 4 * kk + 2 * lg;
    A[kk] = *(const v2f*)(ao + (((size_t)(b * HG_ + lm)) * N_ + n) * D_ + k0);
  }
#pragma unroll
  for (int jt = 0; jt < 4; ++jt) {
    int j = jt * 16 + lm;
    float bv0 = bo[j];
    v8f acc = {};
#pragma unroll
    for (int r = 0; r < 8; ++r) acc[r] = bv0;
#pragma unroll
    for (int kk = 0; kk < 16; ++kk) {
      int k0 = 4 * kk + 2 * lg;
      v2f Bb = *(const v2f*)(Wo + (size_t)j * D_ + k0);
      acc = wmma_f32(A[kk], Bb, acc);
    }
#pragma unroll
    for (int r = 0; r < 8; ++r) {
      int hg = r + 8 * lg;
      out[(((size_t)tok) * HG_ + hg) * D_ + j] = acc[r];
    }
  }
}

// ---------------------------------------------------------------------------
extern "C" void kernel_launch(void* const* d_in, const int* in_sizes, int n_in,
                              void* d_out, int out_size, void* d_ws, size_t ws_size,
                              hipStream_t stream) {
  const float*         x    = (const float*)d_in[0];
  const float*         rot  = (const float*)d_in[1];
  const unsigned char* mask = (const unsigned char*)d_in[2];
  const float* Wq = (const float*)d_in[3];
  const float* bq = (const float*)d_in[4];
  const float* Wk = (const float*)d_in[5];
  const float* bk = (const float*)d_in[6];
  const float* Wv = (const float*)d_in[7];
  const float* bv = (const float*)d_in[8];
  const float* Wo = (const float*)d_in[9];
  const float* bo = (const float*)d_in[10];
  const float* fw = (const float*)d_in[11];

  float* ws = (float*)d_ws;
  float* qe = ws;                                   // [B,HG,N,128] 16 MB
  float* ke = qe + (size_t)B_ * HG_ * N_ * DE_;     // [B,HG,N,128] 16 MB
  float* vT = ke + (size_t)B_ * HG_ * N_ * DE_;     // [B,HG,64,N]   8 MB
  float* wv = vT + (size_t)B_ * HG_ * D_ * N_;      // [B,N,64]    0.5 MB
  float* ao = wv + (size_t)B_ * N_ * 64;            // [B,HG,N,64]   8 MB

  wvec_kernel<<<NT_ / 64, 64, 0, stream>>>(rot, wv);
  qkv_kernel<<<NT_ / 4, 128, 0, stream>>>(x, Wq, bq, Wk, bk, Wv, bv, qe, ke, vT);
  extend_kernel<<<ROWS_, 64, 0, stream>>>(qe, ke, wv, fw);
  dim3 ag(32, 16, 1);
  attn_kernel<<<ag, 128, 0, stream>>>(qe, ke, vT, mask, ao);
  oproj_kernel<<<NT_ / 4, 128, 0, stream>>>(ao, Wo, bo, (float*)d_out);
}